// SpanEmbedder_85392539779538
// MI455X (gfx1250) — compile-verified
//
#include <hip/hip_runtime.h>
#include <hip/hip_bf16.h>
#include <math.h>

#define B_   8
#define S_   2048
#define D_   768
#define NSP  512
#define MAXW 30
#define H_   100
#define NEGV (-1e30f)

typedef __attribute__((ext_vector_type(2))) float v2f;
typedef __attribute__((ext_vector_type(8))) float v8f;

// ---------------------------------------------------------------------------
// Kernel A: att_logits[b*S+s] = dot(seq[b,s,:], att_w) + att_b
// One wave32 per row; lane sums 24 strided elements; shfl_xor tree reduce.
// ---------------------------------------------------------------------------
__global__ __launch_bounds__(256) void span_att_logits_kernel(
    const float* __restrict__ seq, const float* __restrict__ att_w,
    const float* __restrict__ att_b, float* __restrict__ logits) {
  const int lane = threadIdx.x & 31;
  const int wave = threadIdx.x >> 5;
  const int row  = blockIdx.x * 8 + wave;        // [0, B*S)
  const float* p = seq + (size_t)row * D_;
  float acc = 0.f;
#pragma unroll
  for (int k = 0; k < D_ / 32; ++k)
    acc += p[lane + 32 * k] * att_w[lane + 32 * k];
#pragma unroll
  for (int off = 16; off > 0; off >>= 1)
    acc += __shfl_xor(acc, off, 32);
  if (lane == 0) logits[row] = acc + att_b[0];
}

// ---------------------------------------------------------------------------
// Kernel B: per-16-span workgroup (256 threads = 8 waves).
// Stage 1: masked softmax over span width, weighted token sum -> LDS tile
//          (2 spans per wave).
// Stage 2: fp32 WMMA (V_WMMA_F32_16X16X4_F32) GEMM [16x768]x[768x112],
//          one 16-col N-tile per wave (wave 7 idle), bias + tanh,
//          store the 100 valid columns. Loads use clamped addresses so the
//          K-loop is branch-free (dead columns accumulate garbage that is
//          never stored).
// ---------------------------------------------------------------------------
__global__ __launch_bounds__(256) void span_embed_wmma_kernel(
    const float* __restrict__ seq, const int* __restrict__ spi,
    const int* __restrict__ smask, const float* __restrict__ logits,
    const float* __restrict__ ffnn_w, const float* __restrict__ ffnn_b,
    float* __restrict__ out) {
  constexpr int LPAD = D_ + 4;                   // 772: bank-conflict-free columns
  __shared__ float att[16][LPAD];

  const int lane  = threadIdx.x & 31;
  const int wave  = threadIdx.x >> 5;            // 0..7
  const int span0 = blockIdx.x * 16;

  // ---- Stage 1: each wave computes 2 spans ----
  for (int jj = 0; jj < 2; ++jj) {
    const int j  = wave * 2 + jj;                // row in the 16-span tile
    const int s  = span0 + j;
    const int b  = s >> 9;                       // 512 spans per batch
    const int st = spi[2 * s];
    const int en = spi[2 * s + 1];               // exclusive end
    const int width = en - st;                   // in [1, 30]

    // lane w holds logit of token w of the span
    float lg = (lane < width) ? logits[b * S_ + st + lane] : NEGV;
    float mx = lg;
#pragma unroll
    for (int off = 16; off > 0; off >>= 1) mx = fmaxf(mx, __shfl_xor(mx, off, 32));
    float e = (lane < width) ? __expf(lg - mx) : 0.f;
    float sm = e;
#pragma unroll
    for (int off = 16; off > 0; off >>= 1) sm += __shfl_xor(sm, off, 32);
    float attn = e / sm;
    attn *= (smask[s] != 0) ? 1.f : 0.f;         // fold span_mask into weights

    // attended[j][:] = sum_w attn_w * seq[b, st+w, :]
    float acc[D_ / 32];
#pragma unroll
    for (int k = 0; k < D_ / 32; ++k) acc[k] = 0.f;
    const float* rowbase = seq + ((size_t)b * S_ + st) * D_;
    for (int w = 0; w < width; ++w) {
      const float aw = __shfl(attn, w, 32);
      const float* r = rowbase + (size_t)w * D_;
#pragma unroll
      for (int k = 0; k < D_ / 32; ++k) acc[k] += aw * r[lane + 32 * k];
    }
#pragma unroll
    for (int k = 0; k < D_ / 32; ++k) att[j][lane + 32 * k] = acc[k];
  }
  __syncthreads();

  // ---- Stage 2: fp32 WMMA GEMM + tanh, one N-tile per wave ----
  // A frag (16x4 f32): lanes 0-15 -> M=lane, K={0,1}; lanes 16-31 -> M=lane-16, K={2,3}
  // B frag (4x16 f32): lanes 0-15 -> K={0,1}, N=lane; lanes 16-31 -> K={2,3}, N=lane-16
  if (wave < 7) {
    const int half = lane >> 4;                  // 0 or 1
    const int l16  = lane & 15;
    const int n    = wave * 16 + l16;            // output column (feature)
    const bool nok = (n < H_);
    const int nc   = nok ? n : 0;                // clamped: keeps loads in-bounds,
                                                 // dead columns never stored
    v8f c = {0.f, 0.f, 0.f, 0.f, 0.f, 0.f, 0.f, 0.f};
    const float* wp = ffnn_w + nc;

    for (int kk = 0; kk < D_ / 4; ++kk) {        // 192 branch-free K-steps of 4
      const int kb = kk * 4 + 2 * half;
      v2f a, bfr;
      a.x   = att[l16][kb];
      a.y   = att[l16][kb + 1];
      bfr.x = wp[(size_t)kb * H_];
      bfr.y = wp[(size_t)(kb + 1) * H_];
      c = __builtin_amdgcn_wmma_f32_16x16x4_f32(
              /*neg_a=*/false, a, /*neg_b=*/false, bfr,
              /*c_mod=*/(short)0, c, /*reuse_a=*/false, /*reuse_b=*/false);
    }

    const float bias = ffnn_b[nc];
    // C/D layout: lanes 0-15 -> M=vgpr, N=lane; lanes 16-31 -> M=vgpr+8, N=lane-16
#pragma unroll
    for (int r = 0; r < 8; ++r) {
      const int m = r + 8 * half;                // span row in tile
      if (nok)
        out[(size_t)(span0 + m) * H_ + n] = tanhf(c[r] + bias);
    }
  }
}

extern "C" void kernel_launch(void* const* d_in, const int* in_sizes, int n_in,
                              void* d_out, int out_size, void* d_ws, size_t ws_size,
                              hipStream_t stream) {
  const float* seq    = (const float*)d_in[0];   // [B,S,D] f32
  const int*   spi    = (const int*)  d_in[1];   // [B,N,2] i32
  const int*   smask  = (const int*)  d_in[2];   // [B,N]   i32
  const float* att_w  = (const float*)d_in[3];   // [D,1]   f32
  const float* att_b  = (const float*)d_in[4];   // [1]     f32
  const float* ffnn_w = (const float*)d_in[5];   // [D,H]   f32
  const float* ffnn_b = (const float*)d_in[6];   // [H]     f32
  float*       out    = (float*)d_out;           // [B,N,H] f32
  float*       logits = (float*)d_ws;            // B*S floats scratch

  span_att_logits_kernel<<<(B_ * S_) / 8, 256, 0, stream>>>(seq, att_w, att_b, logits);
  span_embed_wmma_kernel<<<(B_ * NSP) / 16, 256, 0, stream>>>(
      seq, spi, smask, logits, ffnn_w, ffnn_b, out);
}